// _DCRProposalLayer_76794015252991
// MI455X (gfx1250) — compile-verified
//
#include <hip/hip_runtime.h>
#include <hip/hip_bf16.h>
#include <stdint.h>

// ---------------------------------------------------------------------------
// DCR proposal layer for MI455X (gfx1250).
// Memory-bound (~370 MB traffic -> ~16us at 23.3 TB/s). No matmul -> no WMMA;
// the CDNA5 feature that matters is the Tensor Data Mover: one
// tensor_load_to_lds per wave streams that wave's 32x81 f32 tile into LDS
// (completion via s_wait_tensorcnt), replacing ~21 per-lane async VMEM ops.
// Top-k = two-pass radix select with an LDS-privatized pass-1 histogram (no sort).
// ---------------------------------------------------------------------------

#define C_TOTAL 81            // classes incl. background
#define RM_ROWS 128           // rows per block in row-max kernel (4 waves)
#define WAVE_ROWS 32          // rows per wave-tile
#define WAVE_ELEMS (WAVE_ROWS * C_TOTAL)   // 2592 f32 per wave TDM tile

typedef uint32_t v4u __attribute__((ext_vector_type(4)));
typedef uint32_t v8u __attribute__((ext_vector_type(8)));

// ws layout (uint32 words):
//   [0, N)                keys (float bits of per-row max; positive -> uint order)
//   [N, N+256)            hist1 (bins on key[31:24])
//   [N+256, N+256+65536)  hist2 (bins on key[23:8] within threshold byte)
//   [N+256+65536, +16)    ctrl:
//     c[0]=b* (top byte)   c[1]=rem1        c[2]=countAbove1
//     c[3]=s* (sub bin)    c[4]=rem2(ties)  c[5]=H* 24-bit threshold prefix
//     c[6]=definite=K-rem2 c[8]=definite slot counter  c[9]=tie counter

__global__ void __launch_bounds__(256) k_zero(uint32_t* p, int n) {
    int i = blockIdx.x * blockDim.x + threadIdx.x;
    if (i < n) p[i] = 0u;
}

// ---- Pass 1: TDM-staged per-row max + fused 256-bin histogram --------------
__global__ void __launch_bounds__(RM_ROWS) k_rowmax(const float* __restrict__ cls,
                                                    uint32_t* __restrict__ keys,
                                                    uint32_t* __restrict__ hist1,
                                                    int N) {
    __shared__ float tile[RM_ROWS * C_TOTAL];   // 41,472 B
    __shared__ uint32_t hblk[256];

    const int tid = threadIdx.x;
    hblk[tid] = 0u;
    hblk[tid + RM_ROWS] = 0u;

    const long long row0 = (long long)blockIdx.x * RM_ROWS;
    const int rows = (int)min((long long)RM_ROWS, (long long)N - row0);

    // --- Tensor Data Mover: each wave DMAs its own 32x81 f32 sub-tile ------
    const uint32_t wave = (uint32_t)(tid >> 5);
    const long long elemBase = (row0 + (long long)wave * WAVE_ROWS) * C_TOTAL;
    const long long remain   = (long long)N * C_TOTAL - elemBase;
    const uint32_t tdim0 = (remain <= 0) ? 0u
                         : (remain >= WAVE_ELEMS ? (uint32_t)WAVE_ELEMS : (uint32_t)remain);
    const uint32_t tile0 = (remain <= 0) ? 0u : (uint32_t)WAVE_ELEMS;  // 0 => TDM NOP
    const uint64_t gaddr = (uint64_t)(cls + elemBase);
    const uint32_t laddr = (uint32_t)(size_t)(&tile[0]) + wave * (WAVE_ELEMS * 4u);

    // D# group 0: count=1 | lds_addr | global_addr[56:0] | type=2
    v4u g0 = (v4u){ 1u,
                    laddr,
                    (uint32_t)gaddr,
                    ((uint32_t)(gaddr >> 32) & 0x01FFFFFFu) | (2u << 30) };
    // D# group 1: data_size=4B; tensor_dim0=tdim0 (OOB zero-fill for tail);
    //             tensor_dim1=1; tile_dim0=2592; tile_dim1=1; dim0_stride=2592
    v8u g1 = (v8u){ (2u << 16),
                    (tdim0 & 0xFFFFu) << 16,
                    (tdim0 >> 16) | (1u << 16),
                    (tile0 & 0xFFFFu) << 16,
                    1u,
                    (uint32_t)WAVE_ELEMS,
                    0u, 0u };

    asm volatile("tensor_load_to_lds %0, %1 th:TH_LOAD_NT"
                 :: "s"(g0), "s"(g1) : "memory");
    asm volatile("s_wait_tensorcnt 0x0" ::: "memory");   // this wave's tile ready
    __syncthreads();                                     // hblk init visibility

    if (tid < rows) {
        const float* r = &tile[tid * C_TOTAL];  // stride 81 -> conflict-free banks
        float m = r[1];
        #pragma unroll
        for (int j = 2; j < C_TOTAL; ++j) m = fmaxf(m, r[j]);
        const uint32_t key = __float_as_uint(m);  // positive float: uint-ordered
        keys[row0 + tid] = key;
        atomicAdd(&hblk[key >> 24], 1u);
    }
    __syncthreads();

    const uint32_t h0 = hblk[tid], h1 = hblk[tid + RM_ROWS];
    if (h0) atomicAdd(&hist1[tid], h0);
    if (h1) atomicAdd(&hist1[tid + RM_ROWS], h1);
}

// ---- Pass 2a: scalar scan of 256 bins (descending) -------------------------
__global__ void k_scan1(const uint32_t* __restrict__ hist1,
                        uint32_t* __restrict__ c, int K) {
    if (threadIdx.x == 0 && blockIdx.x == 0) {
        uint32_t cum = 0; int b = 255;
        for (; b >= 0; --b) {
            uint32_t nb = cum + hist1[b];
            if (nb >= (uint32_t)K) break;
            cum = nb;
        }
        if (b < 0) b = 0;
        c[0] = (uint32_t)b;       // threshold byte
        c[2] = cum;               // strictly-above count
        c[1] = (uint32_t)K - cum; // needed from byte b
    }
}

// ---- Pass 2b: 64K-bin histogram on key[23:8] within threshold byte ---------
__global__ void __launch_bounds__(256) k_hist2(const uint32_t* __restrict__ keys,
                                               const uint32_t* __restrict__ c,
                                               uint32_t* __restrict__ hist2, int N) {
    const uint32_t b = c[0];
    const int stride = gridDim.x * blockDim.x;
    for (int i = blockIdx.x * blockDim.x + threadIdx.x; i < N; i += stride) {
        uint32_t k = keys[i];
        if ((k >> 24) == b) atomicAdd(&hist2[(k >> 8) & 0xFFFFu], 1u);
    }
}

// ---- Pass 2c: parallel descending suffix-scan of 64K bins ------------------
__global__ void __launch_bounds__(256) k_scan2(const uint32_t* __restrict__ hist2,
                                               uint32_t* __restrict__ c, int K) {
    __shared__ uint32_t csum[256];
    __shared__ uint32_t suffix[257];
    const int t = threadIdx.x;
    uint32_t s = 0;
    #pragma unroll 8
    for (int j = 0; j < 256; ++j) s += hist2[t * 256 + j];
    csum[t] = s;
    __syncthreads();
    if (t == 0) {
        uint32_t acc = 0;
        for (int i = 255; i >= 0; --i) { acc += csum[i]; suffix[i] = acc; }
        suffix[256] = 0;
    }
    __syncthreads();
    const uint32_t rem1 = c[1];
    // unique t with suffix[t] >= rem1 > suffix[t+1]
    if (suffix[t] >= rem1 && suffix[t + 1] < rem1) {
        uint32_t cum = suffix[t + 1];
        int sbin = t * 256;
        for (int j = 255; j >= 0; --j) {
            uint32_t nc = cum + hist2[t * 256 + j];
            if (nc >= rem1) { sbin = t * 256 + j; break; }
            cum = nc;
        }
        c[3] = (uint32_t)sbin;
        c[4] = rem1 - cum;                       // ties taken from bin sbin
        c[5] = (c[0] << 16) | (uint32_t)sbin;    // 24-bit threshold prefix
        c[6] = (uint32_t)K - c[4];               // definite count
    }
}

// ---- Pass 3: compact + bbox decode + clip + gather -------------------------
__global__ void __launch_bounds__(256) k_compact(const uint32_t* __restrict__ keys,
                                                 uint32_t* __restrict__ c,
                                                 const float* __restrict__ rois,
                                                 const float* __restrict__ bbox,
                                                 const float* __restrict__ iminfo,
                                                 float* __restrict__ out,
                                                 int* __restrict__ keep_idx,
                                                 int N, int K) {
    const int i = blockIdx.x * blockDim.x + threadIdx.x;
    if (i >= N) return;
    const uint32_t H  = c[5];
    const uint32_t r2 = c[4];
    const uint32_t df = c[6];
    const uint32_t h24 = keys[i] >> 8;

    uint32_t slot; bool sel = false;
    if (h24 > H) {
        slot = atomicAdd(&c[8], 1u);             // [0, definite)
        sel = true;
    } else if (h24 == H) {
        uint32_t t = atomicAdd(&c[9], 1u);       // ties fill [definite, K)
        if (t < r2) { slot = df + t; sel = true; }
    }
    if (!sel) return;

    // boxes = rois[0, i, 1:5]; deltas = bbox_pred[i, 4:8]
    const float x1 = rois[i * 5 + 1], y1 = rois[i * 5 + 2];
    const float x2 = rois[i * 5 + 3], y2 = rois[i * 5 + 4];
    const float4 d = *(const float4*)(bbox + i * 8 + 4);   // 16B aligned
    const float w  = x2 - x1 + 1.0f, h = y2 - y1 + 1.0f;
    const float cx = x1 + 0.5f * w,  cy = y1 + 0.5f * h;
    const float px = d.x * w + cx,   py = d.y * h + cy;
    const float pw = __expf(d.z) * w, ph = __expf(d.w) * h;
    const float Wm1 = iminfo[1] - 1.0f, Hm1 = iminfo[0] - 1.0f;

    float* o = out + (size_t)slot * 5;
    o[0] = 0.0f;
    o[1] = fminf(fmaxf(px - 0.5f * pw, 0.0f), Wm1);
    o[2] = fminf(fmaxf(py - 0.5f * ph, 0.0f), Hm1);
    o[3] = fminf(fmaxf(px + 0.5f * pw, 0.0f), Wm1);
    o[4] = fminf(fmaxf(py + 0.5f * ph, 0.0f), Hm1);
    keep_idx[slot] = i;
}

extern "C" void kernel_launch(void* const* d_in, const int* in_sizes, int n_in,
                              void* d_out, int out_size, void* d_ws, size_t ws_size,
                              hipStream_t stream) {
    const float* rois   = (const float*)d_in[0];   // (1, N, 5)
    const float* cls    = (const float*)d_in[1];   // (N, 81)
    const float* bbox   = (const float*)d_in[2];   // (N, 8)
    const float* iminfo = (const float*)d_in[3];   // (1, 3)

    const int N = in_sizes[1] / C_TOTAL;
    const int K = N / 2;                           // TOP = 0.5

    uint32_t* keys  = (uint32_t*)d_ws;
    uint32_t* hist1 = keys + N;
    uint32_t* hist2 = hist1 + 256;
    uint32_t* ctrl  = hist2 + 65536;
    const int zero_words = 256 + 65536 + 16;

    float* out_blob = (float*)d_out;
    int*   keep_idx = (int*)d_out + (size_t)K * 5;

    k_zero<<<(zero_words + 255) / 256, 256, 0, stream>>>(hist1, zero_words);
    k_rowmax<<<(N + RM_ROWS - 1) / RM_ROWS, RM_ROWS, 0, stream>>>(cls, keys, hist1, N);
    k_scan1<<<1, 32, 0, stream>>>(hist1, ctrl, K);
    k_hist2<<<1024, 256, 0, stream>>>(keys, ctrl, hist2, N);
    k_scan2<<<1, 256, 0, stream>>>(hist2, ctrl, K);
    k_compact<<<(N + 255) / 256, 256, 0, stream>>>(keys, ctrl, rois, bbox, iminfo,
                                                   out_blob, keep_idx, N, K);
}